// TCPGenPrototype_56272661512624
// MI455X (gfx1250) — compile-verified
//
#include <hip/hip_runtime.h>
#include <hip/hip_bf16.h>

// ---------------- problem constants ----------------
constexpr int kB = 4, kT = 128, kU = 32, kC = 64;
constexpr int kV = 5000, kV1 = 5001;
constexpr int kD = 512, kE = 512, kA = 512, kJ = 512;
constexpr int kBT  = kB * kT;        // 512
constexpr int kBU  = kB * kU;        // 128
constexpr int kBUC = kBU * kC;       // 8192
constexpr int kBTU = kB * kT * kU;   // 16384
constexpr float kScale = 0.044194173824159216f;   // 1/sqrt(512)

typedef __attribute__((ext_vector_type(2))) float v2f;
typedef __attribute__((ext_vector_type(8))) float v8f;

// ---------------- zero-fill (ptr_dist is poisoned by the harness) ----------------
__global__ void tcp_zero_f4(float4* __restrict__ p, long long n4) {
  long long i = (long long)blockIdx.x * blockDim.x + threadIdx.x;
  long long stride = (long long)gridDim.x * blockDim.x;
  float4 z = make_float4(0.f, 0.f, 0.f, 0.f);
  for (; i < n4; i += stride) p[i] = z;
}

// ---------------- embedding gathers ----------------
__global__ void tcp_gather_sem(const float* __restrict__ embed,
                               const int* __restrict__ dec,
                               float* __restrict__ sem) {
  int row = blockIdx.x;                    // b*U+u  (128 rows)
  int idx = dec[row];
  const float4* src = (const float4*)(embed + (size_t)idx * kD);
  float4* dst = (float4*)(sem + (size_t)row * kD);
  dst[threadIdx.x] = src[threadIdx.x];     // 128 threads x float4 = 512 floats
}

__global__ void tcp_gather_kv(const float* __restrict__ embed,
                              const float* __restrict__ ookb,
                              const int* __restrict__ masks,
                              float* __restrict__ kv) {
  int row = blockIdx.x;                    // (b*U+u)*C + c  (8192 rows)
  int m = masks[row];
  const float* srow = (m < 0 || m >= kV) ? ookb : (embed + (size_t)m * kD);
  const float4* src = (const float4*)srow;
  float4* dst = (float4*)(kv + (size_t)row * kD);
  dst[threadIdx.x] = src[threadIdx.x];
}

// ---------------- register-blocked fp32 WMMA GEMM ----------------
// C[M,N] = A[M,K] @ W[K,N] + bias.  One wave computes a 64(M) x 32(N) block:
// 4 M-tiles x 2 N-tiles = 8 independent accumulators per K-step, so the 8
// WMMAs pipeline without RAW hazards and vmem:wmma drops to ~1:1.
// grid: blockIdx.x = M/64 block, blockIdx.y = N/32 block.
__global__ __launch_bounds__(32)
void tcp_gemm_f32(const float* __restrict__ A, const float* __restrict__ W,
                  const float* __restrict__ bias, float* __restrict__ Cout,
                  int M, int K, int N) {
  int lane = threadIdx.x;
  int half = lane >> 4;                    // 0: K+{0,1},  1: K+{2,3}
  int ln   = lane & 15;
  int m0 = blockIdx.x * 64;
  int n0 = blockIdx.y * 32;
  const float* Arow[4];
#pragma unroll
  for (int mt = 0; mt < 4; ++mt) Arow[mt] = A + (size_t)(m0 + mt * 16 + ln) * K;
  v8f acc[4][2] = {};
  for (int k = 0; k < K; k += 4) {
    int ka = k + half * 2;
    v2f a[4];
#pragma unroll
    for (int mt = 0; mt < 4; ++mt) a[mt] = *(const v2f*)(Arow[mt] + ka);
    v2f b[2];
#pragma unroll
    for (int nt = 0; nt < 2; ++nt) {
      b[nt].x = W[(size_t)(ka + 0) * N + n0 + nt * 16 + ln];
      b[nt].y = W[(size_t)(ka + 1) * N + n0 + nt * 16 + ln];
    }
#pragma unroll
    for (int mt = 0; mt < 4; ++mt)
#pragma unroll
      for (int nt = 0; nt < 2; ++nt)
        acc[mt][nt] = __builtin_amdgcn_wmma_f32_16x16x4_f32(
            false, a[mt], false, b[nt], (short)0, acc[mt][nt], false, false);
  }
#pragma unroll
  for (int nt = 0; nt < 2; ++nt) {
    int n = n0 + nt * 16 + ln;
    float bv = bias ? bias[n] : 0.f;
#pragma unroll
    for (int mt = 0; mt < 4; ++mt)
#pragma unroll
      for (int r = 0; r < 8; ++r) {
        int om = m0 + mt * 16 + half * 8 + r;   // C/D layout: M split by lane-half
        Cout[(size_t)om * N + n] = acc[mt][nt][r] + bv;
      }
  }
}

// ---------------- attention scores: atten[b,t,u,c] = (qa[b,t]+qs[b,u]) . keys[b,u,c] * scale ----
// One wave computes 64(T) x 32(C) per (b,u); qs fragment hoisted (shared by all M-tiles).
// grid: x = C/32, y = T/64, z = b*U+u
__global__ __launch_bounds__(32)
void tcp_score(const float* __restrict__ qa, const float* __restrict__ qs,
               const float* __restrict__ keys, float* __restrict__ atten) {
  int bu = blockIdx.z;
  int b = bu / kU, u = bu % kU;
  int lane = threadIdx.x, half = lane >> 4, ln = lane & 15;
  int t0 = blockIdx.y * 64;
  int c0 = blockIdx.x * 32;
  const float* qa_b   = qa   + (size_t)b  * kT * kA;
  const float* qs_bu  = qs   + (size_t)bu * kA;
  const float* key_bu = keys + (size_t)bu * kC * kA;
  v8f acc[4][2] = {};
  for (int k = 0; k < kA; k += 4) {
    int ka = k + half * 2;
    v2f sv = *(const v2f*)(qs_bu + ka);
    v2f a[4];
#pragma unroll
    for (int mt = 0; mt < 4; ++mt) {
      v2f qv = *(const v2f*)(qa_b + (size_t)(t0 + mt * 16 + ln) * kA + ka);
      a[mt].x = qv.x + sv.x;                       // fused q = qa + qs
      a[mt].y = qv.y + sv.y;
    }
    v2f bf[2];
#pragma unroll
    for (int nt = 0; nt < 2; ++nt)                 // keys row-major in K => v2f load
      bf[nt] = *(const v2f*)(key_bu + (size_t)(c0 + nt * 16 + ln) * kA + ka);
#pragma unroll
    for (int mt = 0; mt < 4; ++mt)
#pragma unroll
      for (int nt = 0; nt < 2; ++nt)
        acc[mt][nt] = __builtin_amdgcn_wmma_f32_16x16x4_f32(
            false, a[mt], false, bf[nt], (short)0, acc[mt][nt], false, false);
  }
#pragma unroll
  for (int nt = 0; nt < 2; ++nt) {
    int c = c0 + nt * 16 + ln;
#pragma unroll
    for (int mt = 0; mt < 4; ++mt)
#pragma unroll
      for (int r = 0; r < 8; ++r) {
        int tm = t0 + mt * 16 + half * 8 + r;
        atten[((size_t)(b * kT + tm) * kU + u) * kC + c] = acc[mt][nt][r] * kScale;
      }
  }
}

// ---------------- masked softmax over C=64, one wave per (b,t,u) row ----------------
__global__ __launch_bounds__(32)
void tcp_softmax(float* __restrict__ atten, const int* __restrict__ masks) {
  int row = blockIdx.x;                    // (b*T+t)*U+u
  int u = row % kU;
  int b = row / (kT * kU);
  int lane = threadIdx.x;
  const int* mrow = masks + ((size_t)b * kU + u) * kC;
  float* arow = atten + (size_t)row * kC;
  int m0 = mrow[lane], m1 = mrow[lane + 32];
  float v0 = (m0 == -1) ? -1e9f : arow[lane];
  float v1 = (m1 == -1) ? -1e9f : arow[lane + 32];
  float mx = fmaxf(v0, v1);
  for (int off = 16; off; off >>= 1) mx = fmaxf(mx, __shfl_xor(mx, off, 32));
  float e0 = __expf(v0 - mx), e1 = __expf(v1 - mx);
  float s = e0 + e1;
  for (int off = 16; off; off >>= 1) s += __shfl_xor(s, off, 32);
  float inv = 1.0f / s;
  arow[lane]      = e0 * inv;
  arow[lane + 32] = e1 * inv;
}

// ---------------- x[b,t,u,:] = atten[b,t,u,:] @ kv[b,u,:,:]  (M=T,K=C,N=D per bu) ------
// One wave computes 64(T) x 32(D) per (b,u).  grid: x = D/32, y = T/64, z = b*U+u
__global__ __launch_bounds__(32)
void tcp_xgemm(const float* __restrict__ atten, const float* __restrict__ kv,
               float* __restrict__ x) {
  int bu = blockIdx.z;
  int b = bu / kU, u = bu % kU;
  int lane = threadIdx.x, half = lane >> 4, ln = lane & 15;
  int t0 = blockIdx.y * 64;
  int n0 = blockIdx.x * 32;
  const float* kv_bu = kv + (size_t)bu * kC * kD;
  v8f acc[4][2] = {};
  for (int k = 0; k < kC; k += 4) {
    int ka = k + half * 2;
    v2f a[4];
#pragma unroll
    for (int mt = 0; mt < 4; ++mt)                 // atten contiguous in K=C => v2f load
      a[mt] = *(const v2f*)(atten + ((size_t)(b * kT + t0 + mt * 16 + ln) * kU + u) * kC + ka);
    v2f bf[2];
#pragma unroll
    for (int nt = 0; nt < 2; ++nt) {
      bf[nt].x = kv_bu[(size_t)(ka + 0) * kD + n0 + nt * 16 + ln];
      bf[nt].y = kv_bu[(size_t)(ka + 1) * kD + n0 + nt * 16 + ln];
    }
#pragma unroll
    for (int mt = 0; mt < 4; ++mt)
#pragma unroll
      for (int nt = 0; nt < 2; ++nt)
        acc[mt][nt] = __builtin_amdgcn_wmma_f32_16x16x4_f32(
            false, a[mt], false, bf[nt], (short)0, acc[mt][nt], false, false);
  }
#pragma unroll
  for (int nt = 0; nt < 2; ++nt) {
    int n = n0 + nt * 16 + ln;
#pragma unroll
    for (int mt = 0; mt < 4; ++mt)
#pragma unroll
      for (int r = 0; r < 8; ++r) {
        int tm = t0 + mt * 16 + half * 8 + r;
        x[((size_t)(b * kT + tm) * kU + u) * kD + n] = acc[mt][nt][r];
      }
  }
}

// ---------------- scatter atten into ptr_dist; one thread per (b,t,u) row --------------
// Intra-row index collisions serialize inside the owning thread => no atomics needed.
// Prefetch (write-intent) the slot 4 iterations ahead to hide the RMW latency chain
// (lowers to gfx1250 global_prefetch_b8).
__global__ void tcp_scatter(const float* __restrict__ atten,
                            const int* __restrict__ masks,
                            float* __restrict__ ptr) {
  int row = blockIdx.x * blockDim.x + threadIdx.x;   // (b*T+t)*U+u
  if (row >= kBTU) return;
  int u = row % kU;
  int b = row / (kT * kU);
  const int* mrow = masks + ((size_t)b * kU + u) * kC;
  const float* arow = atten + (size_t)row * kC;
  float* prow = ptr + (size_t)row * kV1;
  for (int c = 0; c < kC; ++c) {
    if (c + 4 < kC) {
      int mp = mrow[c + 4];
      if (mp >= 0) __builtin_prefetch(prow + mp, 1, 3);
    }
    int m = mrow[c];
    if (m >= 0) prow[m] += arow[c];                  // pad (-1) dropped; m==V keeps ooKB slot
  }
}

extern "C" void kernel_launch(void* const* d_in, const int* in_sizes, int n_in,
                              void* d_out, int out_size, void* d_ws, size_t ws_size,
                              hipStream_t stream) {
  (void)in_sizes; (void)n_in; (void)out_size; (void)ws_size;
  const float* encoder_out = (const float*)d_in[0];
  const int*   decoder_in  = (const int*)  d_in[1];
  const int*   masks_mat   = (const int*)  d_in[2];
  const float* embed_w     = (const float*)d_in[3];
  const float* ookb_w      = (const float*)d_in[4];
  const float* qa_w        = (const float*)d_in[5];
  const float* qa_b        = (const float*)d_in[6];
  const float* qs_w        = (const float*)d_in[7];
  const float* qs_b        = (const float*)d_in[8];
  const float* k_w         = (const float*)d_in[9];
  const float* k_b         = (const float*)d_in[10];
  const float* db_w        = (const float*)d_in[11];
  const float* db_b        = (const float*)d_in[12];

  float* out      = (float*)d_out;
  float* ptr_dist = out;                                   // [B,T,U,V1]
  float* h_ptr    = ptr_dist + (size_t)kBTU * kV1;         // [B,T,U,D]
  float* dbias    = h_ptr + (size_t)kBTU * kD;             // [B,T,U,J]

  float* ws    = (float*)d_ws;
  float* qa    = ws;                                       // [BT, A]
  float* qs    = qa  + (size_t)kBT  * kA;                  // [BU, A]
  float* sem   = qs  + (size_t)kBU  * kA;                  // [BU, D]
  float* kv    = sem + (size_t)kBU  * kD;                  // [BUC, D]
  float* keys  = kv  + (size_t)kBUC * kD;                  // [BUC, A]
  float* atten = keys + (size_t)kBUC * kA;                 // [BTU, C]

  // 1) zero the sparse pointer-distribution output (327 MB, HBM-roofline bound)
  long long n4 = (long long)kBTU * kV1 / 4;
  tcp_zero_f4<<<4096, 256, 0, stream>>>((float4*)ptr_dist, n4);

  // 2) embedding gathers
  tcp_gather_sem<<<kBU, 128, 0, stream>>>(embed_w, decoder_in, sem);
  tcp_gather_kv<<<kBUC, 128, 0, stream>>>(embed_w, ookb_w, masks_mat, kv);

  // 3) dense projections (register-blocked fp32 WMMA, 64x32 per wave)
  tcp_gemm_f32<<<dim3(kBT / 64,  kA / 32), 32, 0, stream>>>(encoder_out, qa_w, qa_b, qa,   kBT,  kE, kA);
  tcp_gemm_f32<<<dim3(kBU / 64,  kA / 32), 32, 0, stream>>>(sem,         qs_w, qs_b, qs,   kBU,  kD, kA);
  tcp_gemm_f32<<<dim3(kBUC / 64, kA / 32), 32, 0, stream>>>(kv,          k_w,  k_b,  keys, kBUC, kD, kA);

  // 4) attention: scores (fused q=qa+qs), masked softmax, value contraction
  tcp_score  <<<dim3(kC / 32, kT / 64, kBU), 32, 0, stream>>>(qa, qs, keys, atten);
  tcp_softmax<<<kBTU, 32, 0, stream>>>(atten, masks_mat);
  tcp_xgemm  <<<dim3(kD / 32, kT / 64, kBU), 32, 0, stream>>>(atten, kv, h_ptr);

  // 5) dbias = x @ db_w + db_b  (largest GEMM: 16384x512x512)
  tcp_gemm_f32<<<dim3(kBTU / 64, kJ / 32), 32, 0, stream>>>(h_ptr, db_w, db_b, dbias, kBTU, kD, kJ);

  // 6) scatter attention weights into ptr_dist
  tcp_scatter<<<kBTU / 256, 256, 0, stream>>>(atten, masks_mat, ptr_dist);
}